// StackedLinear_15771119911329
// MI455X (gfx1250) — compile-verified
//
#include <hip/hip_runtime.h>

typedef __attribute__((ext_vector_type(2))) float v2f;
typedef __attribute__((ext_vector_type(4))) float v4f;
typedef __attribute__((ext_vector_type(8))) float v8f;

#define IN_F    1024
#define OUT_F   1024
#define NEXP    16
#define NBATCH  8192
#define BM      64
#define BN      64
#define BK      32
#define LSTRIDE 36            // floats; 144B rows: 16B-aligned stores, all-64-bank fragment reads
#define LBUF    (BM * LSTRIDE)

// CDNA5 async global->LDS copy (ASYNCcnt-tracked); flip to 0 to fall back to
// the register-staged double-buffer pipeline.
#define USE_ASYNC_LDS 1

#if USE_ASYNC_LDS
__device__ __forceinline__ void async_copy_b128(uint32_t lds_byte_off, const float* gptr) {
    // VDST = per-lane LDS byte address, VADDR = per-lane 64-bit global address.
    asm volatile("global_load_async_to_lds_b128 %0, %1, off"
                 :: "v"(lds_byte_off), "v"(gptr)
                 : "memory");
}
__device__ __forceinline__ void wait_async0() {
    asm volatile("s_wait_asynccnt 0x0" ::: "memory");
}
__device__ __forceinline__ uint32_t lds_off(const void* p) {
    // flat->LDS: low 32 bits of a flat shared-memory address are the LDS byte offset
    return (uint32_t)(uintptr_t)p;
}
#endif

__global__ void zero_counts(int* cnt) {
    if (threadIdx.x < NEXP) cnt[threadIdx.x] = 0;
}

__global__ void bucket_rows(const long long* __restrict__ idx, int* cnt, int* perm) {
    int b = blockIdx.x * blockDim.x + threadIdx.x;
    if (b < NBATCH) {
        int e = (int)idx[b];
        int pos = atomicAdd(&cnt[e], 1);
        perm[e * NBATCH + pos] = b;
    }
}

__global__ void __launch_bounds__(256)
expert_gemm(const float* __restrict__ X, const float* __restrict__ W,
            const float* __restrict__ bias, const int* __restrict__ cnt,
            const int* __restrict__ perm, float* __restrict__ out) {
    const int e  = blockIdx.z;
    const int mt = blockIdx.y;
    const int nb = blockIdx.x * BN;
    const int cnt_e = cnt[e];
    if (mt * BM >= cnt_e) return;             // 16x overdispatch; most blocks exit here

    __shared__ float As[2][LBUF];
    __shared__ float Bs[2][LBUF];
    __shared__ int   rows_s[BM];

    const int tid = threadIdx.x;
    if (tid < BM) {
        int gm = mt * BM + tid;
        rows_s[tid] = (gm < cnt_e) ? perm[e * NBATCH + gm] : 0;   // row 0 = safe dummy
    }
    __syncthreads();

    const int lane = tid & 31;
    const int wave = tid >> 5;
    const int wm   = wave & 3;     // M sub-tile (16 rows)
    const int wn   = wave >> 2;    // N half (32 cols)
    const int ml   = lane & 15;
    const int kq   = lane >> 4;    // 0/1: K-pair select per WMMA A/B layout

    // -------- loop-invariant staging pointers --------
    const int r0 = tid >> 3;       // 0..31: first row this thread stages
    const int c4 = tid & 7;        // float4 column within the 32-wide K chunk
    const int wrow_base = e * OUT_F + nb;

    const float* xpA = X + (size_t)rows_s[r0]      * IN_F + c4 * 4;
    const float* xpB = X + (size_t)rows_s[r0 + 32] * IN_F + c4 * 4;
    const float* wpA = W + (size_t)(wrow_base + r0)      * IN_F + c4 * 4;
    const float* wpB = W + (size_t)(wrow_base + r0 + 32) * IN_F + c4 * 4;

    const int ldsOffA = r0 * LSTRIDE + c4 * 4;
    const int ldsOffB = (r0 + 32) * LSTRIDE + c4 * 4;

#if USE_ASYNC_LDS
    const uint32_t aOff[2] = { lds_off(&As[0][ldsOffA]), lds_off(&As[1][ldsOffA]) };
    const uint32_t bOff[2] = { lds_off(&As[0][ldsOffB]), lds_off(&As[1][ldsOffB]) };
    const uint32_t cOff[2] = { lds_off(&Bs[0][ldsOffA]), lds_off(&Bs[1][ldsOffA]) };
    const uint32_t dOff[2] = { lds_off(&Bs[0][ldsOffB]), lds_off(&Bs[1][ldsOffB]) };

    // -------- prologue: async-stage chunk 0 into buffer 0 --------
    async_copy_b128(aOff[0], xpA);
    async_copy_b128(bOff[0], xpB);
    async_copy_b128(cOff[0], wpA);
    async_copy_b128(dOff[0], wpB);
    wait_async0();
    __syncthreads();
#else
    {
        v4f xa = *(const v4f*)xpA;
        v4f xb = *(const v4f*)xpB;
        v4f wa = *(const v4f*)wpA;
        v4f wb = *(const v4f*)wpB;
        *(v4f*)&As[0][ldsOffA] = xa;
        *(v4f*)&As[0][ldsOffB] = xb;
        *(v4f*)&Bs[0][ldsOffA] = wa;
        *(v4f*)&Bs[0][ldsOffB] = wb;
    }
    __syncthreads();
#endif

    v8f c0 = {};   // N cols [wn*32 +  0, +16)
    v8f c1 = {};   // N cols [wn*32 + 16, +32)

    int cur = 0;
    for (int k0 = 0; k0 < IN_F; k0 += BK) {
        const bool more = (k0 + BK) < IN_F;

#if USE_ASYNC_LDS
        // Kick next-chunk async copies into the alternate buffer; the async
        // engine fills LDS while this wave issues WMMAs from the current one.
        if (more) {
            const int nxt = cur ^ 1;
            async_copy_b128(aOff[nxt], xpA + k0 + BK);
            async_copy_b128(bOff[nxt], xpB + k0 + BK);
            async_copy_b128(cOff[nxt], wpA + k0 + BK);
            async_copy_b128(dOff[nxt], wpB + k0 + BK);
        }
#else
        v4f nxa, nxb, nwa, nwb;
        if (more) {
            nxa = *(const v4f*)(xpA + k0 + BK);
            nxb = *(const v4f*)(xpB + k0 + BK);
            nwa = *(const v4f*)(wpA + k0 + BK);
            nwb = *(const v4f*)(wpB + k0 + BK);
        }
#endif

        const float* Ab = As[cur];
        const float* Bb = Bs[cur];
        #pragma unroll
        for (int kk = 0; kk < BK; kk += 4) {
            const v2f a  = *(const v2f*)&Ab[(wm * 16 + ml)      * LSTRIDE + kk + kq * 2];
            const v2f b0 = *(const v2f*)&Bb[(wn * 32 + ml)      * LSTRIDE + kk + kq * 2];
            const v2f b1 = *(const v2f*)&Bb[(wn * 32 + 16 + ml) * LSTRIDE + kk + kq * 2];
            c0 = __builtin_amdgcn_wmma_f32_16x16x4_f32(false, a, false, b0, (short)0, c0, false, false);
            c1 = __builtin_amdgcn_wmma_f32_16x16x4_f32(false, a, false, b1, (short)0, c1, false, false);
        }

#if USE_ASYNC_LDS
        wait_async0();       // own async copies into buf^1 complete
#else
        if (more) {
            const int nxt = cur ^ 1;
            *(v4f*)&As[nxt][ldsOffA] = nxa;
            *(v4f*)&As[nxt][ldsOffB] = nxb;
            *(v4f*)&Bs[nxt][ldsOffA] = nwa;
            *(v4f*)&Bs[nxt][ldsOffB] = nwb;
        }
#endif
        __syncthreads();     // single barrier per chunk (double-buffered)
        cur ^= 1;
    }

    // -------- epilogue: bias add + scatter store by gathered row id --------
    // C/D layout: VGPR v, lanes 0-15 -> M=v, lanes 16-31 -> M=v+8; N = lane&15.
    const int   n0    = nb + wn * 32 + ml;
    const float bias0 = bias[e * OUT_F + n0];
    const float bias1 = bias[e * OUT_F + n0 + 16];

    #pragma unroll
    for (int v = 0; v < 8; ++v) {
        int mloc = wm * 16 + v + kq * 8;
        if (mt * BM + mloc < cnt_e) {
            int row = rows_s[mloc];
            out[(size_t)row * OUT_F + n0]      = c0[v] + bias0;
            out[(size_t)row * OUT_F + n0 + 16] = c1[v] + bias1;
        }
    }
}

extern "C" void kernel_launch(void* const* d_in, const int* in_sizes, int n_in,
                              void* d_out, int out_size, void* d_ws, size_t ws_size,
                              hipStream_t stream) {
    const float*     X   = (const float*)d_in[0];
    const long long* idx = (const long long*)d_in[1];   // int64 indices
    const float*     W   = (const float*)d_in[2];
    const float*     B   = (const float*)d_in[3];
    float* out = (float*)d_out;

    int* cnt  = (int*)d_ws;                       // 16 counters
    int* perm = (int*)((char*)d_ws + 256);        // 16 * 8192 row ids (~512 KB)

    zero_counts<<<1, 32, 0, stream>>>(cnt);
    bucket_rows<<<NBATCH / 256, 256, 0, stream>>>(idx, cnt, perm);

    dim3 grid(OUT_F / BN, NBATCH / BM, NEXP);
    expert_gemm<<<grid, 256, 0, stream>>>(X, W, B, cnt, perm, out);
}